// VariationalGCNEncoder_7413113553702
// MI455X (gfx1250) — compile-verified
//
#include <hip/hip_runtime.h>
#include <hip/hip_bf16.h>
#include <math.h>

typedef __attribute__((ext_vector_type(2))) float v2f;
typedef __attribute__((ext_vector_type(8))) float v8f;

#define HID 128
#define FOUT 64
#define FIN 256

__device__ __forceinline__ void atomic_add_f32(float* p, float v) {
  __hip_atomic_fetch_add(p, v, __ATOMIC_RELAXED, __HIP_MEMORY_SCOPE_AGENT);
}

// ---------------- utility kernels ----------------

__global__ void zero_f32(float* __restrict__ p, size_t count) {
  size_t i = (size_t)blockIdx.x * blockDim.x + threadIdx.x;
  if (i < count) p[i] = 0.0f;
}

__global__ void init_deg(float* __restrict__ deg, int n) {
  int i = blockIdx.x * blockDim.x + threadIdx.x;
  if (i < n) deg[i] = 1.0f;  // self-loop contributes 1
}

__global__ void deg_accum(const long long* __restrict__ dst, float* __restrict__ deg, int E) {
  int e = blockIdx.x * blockDim.x + threadIdx.x;
  if (e < E) atomic_add_f32(&deg[(int)dst[e]], 1.0f);
}

__global__ void compute_dinv(const float* __restrict__ deg, float* __restrict__ dinv, int n) {
  int i = blockIdx.x * blockDim.x + threadIdx.x;
  if (i < n) {
    float d = deg[i];
    dinv[i] = (d > 0.0f) ? rsqrtf(d) : 0.0f;
  }
}

// ---------------- fp32 WMMA GEMM ----------------
// C[M x N] = A[M x K] * B[K x N]; row-major. One wave32 per 16x16 tile.
// Uses V_WMMA_F32_16X16X4_F32 (D = A(16x4) * B(4x16) + C).
// Fragment layouts per CDNA5 ISA 7.12.2:
//   A 16x4 : lane l holds row M=l%16, VGPR v holds K = v + 2*(l/16)
//   B 4x16 : lane l holds col N=l%16, VGPR v holds K = v + 2*(l/16)
//   C 16x16: lane l holds col N=l%16, VGPR v holds row M = v + 8*(l/16)
__global__ __launch_bounds__(32) void wmma_gemm_f32(
    const float* __restrict__ A, int lda,
    const float* __restrict__ B, int ldb,
    float* __restrict__ C, int ldc,
    int K) {
  const int lane   = threadIdx.x;        // 0..31
  const int mIdx   = lane & 15;          // row within tile (A) / col within tile (B,C)
  const int laneHi = lane >> 4;          // 0 or 1
  const int kOff   = laneHi * 2;

  const int rowBase = blockIdx.x * 16;
  const int colBase = blockIdx.y * 16;

  const float* __restrict__ Arow = A + (size_t)(rowBase + mIdx) * lda;
  v8f acc = {};

  for (int k = 0; k < K; k += 4) {
    v2f a, b;
    a.x = Arow[k + kOff];
    a.y = Arow[k + kOff + 1];
    b.x = B[(size_t)(k + kOff) * ldb + colBase + mIdx];
    b.y = B[(size_t)(k + kOff + 1) * ldb + colBase + mIdx];
    acc = __builtin_amdgcn_wmma_f32_16x16x4_f32(
        /*neg_a=*/false, a, /*neg_b=*/false, b,
        /*c_mod=*/(short)0, acc, /*reuse_a=*/false, /*reuse_b=*/false);
  }

  const int crow0 = rowBase + laneHi * 8;
  const int ccol  = colBase + mIdx;
#pragma unroll
  for (int v = 0; v < 8; ++v) {
    C[(size_t)(crow0 + v) * ldc + ccol] = acc[v];
  }
}

// ---------------- layer 1 edge scatter ----------------
// out1[dst, f] += h[src, f] * dinv[src] * dinv[dst]   (128 feats per edge)
__global__ __launch_bounds__(HID) void scatter_l1(
    const long long* __restrict__ src, const long long* __restrict__ dst,
    const float* __restrict__ h, const float* __restrict__ dinv,
    float* __restrict__ out1, int E) {
  int e = blockIdx.x;
  if (e >= E) return;
  int f = threadIdx.x;
  int s = (int)src[e];
  int d = (int)dst[e];
  float c = dinv[s] * dinv[d];
  atomic_add_f32(&out1[(size_t)d * HID + f], h[(size_t)s * HID + f] * c);
}

// self-loop + bias + ReLU + row L2-normalize; in-place on out1 -> h_norm
__global__ __launch_bounds__(HID) void post_l1(
    float* __restrict__ out1, const float* __restrict__ h,
    const float* __restrict__ dinv, const float* __restrict__ b1) {
  __shared__ float red[HID];
  int i = blockIdx.x;
  int f = threadIdx.x;
  float di = dinv[i];
  float v = out1[(size_t)i * HID + f] + h[(size_t)i * HID + f] * (di * di) + b1[f];
  v = fmaxf(v, 0.0f);                 // ReLU
  red[f] = v * v;
  __syncthreads();
  for (int s = HID / 2; s > 0; s >>= 1) {
    if (f < s) red[f] += red[f + s];
    __syncthreads();
  }
  float nrm = sqrtf(red[0]);
  out1[(size_t)i * HID + f] = v / fmaxf(nrm, 1e-12f);
}

// ---------------- layer 2 edge scatter ----------------
// hm2 holds [h_norm*W_mu | h_norm*W_ls] as N x 128 (mu cols 0..63, ls cols 64..127)
// d_out = [mu (n*64) | logstd (n*64)]
__global__ __launch_bounds__(HID) void scatter_l2(
    const long long* __restrict__ src, const long long* __restrict__ dst,
    const float* __restrict__ hm2, const float* __restrict__ dinv,
    float* __restrict__ out, int E, int n) {
  int e = blockIdx.x;
  if (e >= E) return;
  int f = threadIdx.x;
  int s = (int)src[e];
  int d = (int)dst[e];
  float c = dinv[s] * dinv[d];
  float val = hm2[(size_t)s * HID + f] * c;
  float* p = (f < FOUT)
                 ? &out[(size_t)d * FOUT + f]
                 : &out[(size_t)n * FOUT + (size_t)d * FOUT + (f - FOUT)];
  atomic_add_f32(p, val);
}

__global__ __launch_bounds__(HID) void post_l2(
    float* __restrict__ out, const float* __restrict__ hm2,
    const float* __restrict__ dinv, const float* __restrict__ b_mu,
    const float* __restrict__ b_ls, int n) {
  int i = blockIdx.x;
  int f = threadIdx.x;
  float di2 = dinv[i] * dinv[i];
  float sl = hm2[(size_t)i * HID + f] * di2;
  if (f < FOUT) {
    out[(size_t)i * FOUT + f] += sl + b_mu[f];
  } else {
    out[(size_t)n * FOUT + (size_t)i * FOUT + (f - FOUT)] += sl + b_ls[f - FOUT];
  }
}

// ---------------- host launcher ----------------

extern "C" void kernel_launch(void* const* d_in, const int* in_sizes, int n_in,
                              void* d_out, int out_size, void* d_ws, size_t ws_size,
                              hipStream_t stream) {
  const float*     x    = (const float*)d_in[0];
  const long long* ei   = (const long long*)d_in[1];  // int64 [2, E]
  const float*     W1   = (const float*)d_in[2];
  const float*     b1   = (const float*)d_in[3];
  const float*     W_mu = (const float*)d_in[4];
  const float*     b_mu = (const float*)d_in[5];
  const float*     W_ls = (const float*)d_in[6];
  const float*     b_ls = (const float*)d_in[7];
  float*           out  = (float*)d_out;

  const int n = in_sizes[0] / FIN;       // 50000
  const int E = in_sizes[1] / 2;         // 800000
  const long long* src = ei;
  const long long* dst = ei + E;

  // workspace layout (floats): deg[n] | dinv[n] | bufH[n*128] | bufO[n*128]
  float* wsf  = (float*)d_ws;
  float* deg  = wsf;
  float* dinv = wsf + n;
  float* bufH = wsf + 2 * (size_t)n;                     // h, later hm2
  float* bufO = bufH + (size_t)n * HID;                  // out1 / h_norm

  const size_t nh = (size_t)n * HID;

  // 1) degrees and dinv
  init_deg<<<(n + 255) / 256, 256, 0, stream>>>(deg, n);
  deg_accum<<<(E + 255) / 256, 256, 0, stream>>>(dst, deg, E);
  compute_dinv<<<(n + 255) / 256, 256, 0, stream>>>(deg, dinv, n);

  // 2) h = x @ W1   [n x 256] * [256 x 128]
  {
    dim3 grid(n / 16, HID / 16);
    wmma_gemm_f32<<<grid, 32, 0, stream>>>(x, FIN, W1, HID, bufH, HID, FIN);
  }

  // 3) layer-1 aggregation
  zero_f32<<<(unsigned)((nh + 255) / 256), 256, 0, stream>>>(bufO, nh);
  scatter_l1<<<E, HID, 0, stream>>>(src, dst, bufH, dinv, bufO, E);
  post_l1<<<n, HID, 0, stream>>>(bufO, bufH, dinv, b1);   // bufO now = h_norm

  // 4) hm2 = [h_norm @ W_mu | h_norm @ W_ls]  written into bufH (n x 128)
  {
    dim3 grid(n / 16, FOUT / 16);
    wmma_gemm_f32<<<grid, 32, 0, stream>>>(bufO, HID, W_mu, FOUT, bufH, HID, HID);
    wmma_gemm_f32<<<grid, 32, 0, stream>>>(bufO, HID, W_ls, FOUT, bufH + FOUT, HID, HID);
  }

  // 5) layer-2 aggregation directly into d_out
  zero_f32<<<(unsigned)(((size_t)out_size + 255) / 256), 256, 0, stream>>>(out, (size_t)out_size);
  scatter_l2<<<E, HID, 0, stream>>>(src, dst, bufH, dinv, out, E, n);
  post_l2<<<n, HID, 0, stream>>>(out, bufH, dinv, b_mu, b_ls, n);
}